// Volume_49100066127892
// MI455X (gfx1250) — compile-verified
//
#include <hip/hip_runtime.h>
#include <hip/hip_bf16.h>

// Problem constants (stage 0 only: VOLUME_DIMS=[96], scale=1)
#define VD   96
#define IMH  480
#define IMW  640
#define NC   16
#define NV   6
static constexpr int NP = VD * VD * VD;          // 884736 voxels

typedef float v2f __attribute__((ext_vector_type(2)));
typedef float v8f __attribute__((ext_vector_type(8)));
// 4-byte-aligned float2 for 64-bit row-pair gathers at arbitrary column parity
typedef float v2fu __attribute__((ext_vector_type(2), aligned(4)));

// ---------------------------------------------------------------------------
// Setup: Pm[v] = intrs[v] (3x4) @ inv(c2w[v]) (4x4)  -> 6 * 12 floats in d_ws
// c2w is rigid: inv = [R^T | -R^T t ; 0 0 0 1]
// ---------------------------------------------------------------------------
__global__ void proj_setup_kernel(const float* __restrict__ intrs,
                                  const float* __restrict__ c2ws,
                                  float* __restrict__ Pm) {
    int v = threadIdx.x;
    if (v >= NV) return;
    const float* M = c2ws + v * 16;   // 4x4 row-major
    const float* K = intrs + v * 12;  // 3x4 row-major
    float w2c[4][4];
    #pragma unroll
    for (int i = 0; i < 3; ++i)
        #pragma unroll
        for (int j = 0; j < 3; ++j)
            w2c[i][j] = M[j * 4 + i];                    // R^T
    #pragma unroll
    for (int i = 0; i < 3; ++i) {
        float acc = 0.0f;
        #pragma unroll
        for (int j = 0; j < 3; ++j) acc += M[j * 4 + i] * M[j * 4 + 3];
        w2c[i][3] = -acc;                                // -R^T t
    }
    w2c[3][0] = 0.0f; w2c[3][1] = 0.0f; w2c[3][2] = 0.0f; w2c[3][3] = 1.0f;
    #pragma unroll
    for (int i = 0; i < 3; ++i)
        #pragma unroll
        for (int c = 0; c < 4; ++c) {
            float acc = 0.0f;
            #pragma unroll
            for (int k = 0; k < 4; ++k) acc += K[i * 4 + k] * w2c[k][c];
            Pm[v * 12 + i * 4 + c] = acc;
        }
}

// ---------------------------------------------------------------------------
// Main kernel. 256 threads = 8 waves; each wave computes 16 voxels with
// lane pairs (l, l+16) splitting the 6 views 3/3. Projections for all views
// are produced by two v_wmma_f32_16x16x4_f32 ops:
//   D = S (16x4 stacked projection rows) x B (4x16 voxel homogeneous coords)
// Row packing of S1/S2 is chosen so each lane's D VGPRs hold exactly its
// 3 views' (u*z, v*z, z) triples with no cross-lane shuffle:
//   S1 rows 0-5   = views 0,1 (u,v,w)   -> lanes 0-15,  d1[0..5]
//   S1 rows 8-13  = views 2,3           -> lanes 16-31, d1[0..5]
//   S2 rows 0-2   = view 4              -> lanes 0-15,  d2[0..2]
//   S2 rows 8-10  = view 5              -> lanes 16-31, d2[0..2]
// ---------------------------------------------------------------------------
__device__ __forceinline__ float s1_elem(const float* Pm, int row, int k) {
    if (row < 6)                return Pm[(row / 3) * 12 + (row % 3) * 4 + k];
    if (row >= 8 && row < 14)   return Pm[(2 + (row - 8) / 3) * 12 + ((row - 8) % 3) * 4 + k];
    return 0.0f;
}
__device__ __forceinline__ float s2_elem(const float* Pm, int row, int k) {
    if (row < 3)                return Pm[4 * 12 + row * 4 + k];
    if (row >= 8 && row < 11)   return Pm[5 * 12 + (row - 8) * 4 + k];
    return 0.0f;
}

__global__ __launch_bounds__(256) void volume_kernel(
        const float* __restrict__ feat,   // (6,16,480,640)
        const float* __restrict__ Pm,     // (6,3,4) projection matrices
        float* __restrict__ out) {        // mean(16,P) | var(16,P) | mask(P)
    const int lane = threadIdx.x & 31;
    const int warp = threadIdx.x >> 5;
    const int half = lane >> 4;       // 0: views {0,1,4}; 1: views {2,3,5}
    const int col  = lane & 15;       // voxel slot in this wave's 16-voxel tile
    const int p    = (blockIdx.x * 8 + warp) * 16 + col;

    // voxel world coords, meshgrid 'ij': p = ((ix*D)+iy)*D+iz
    const int izv = p % VD;
    const int iyv = (p / VD) % VD;
    const int ixv = p / (VD * VD);
    const float step = 2.0f / (float)(VD - 1);
    const float wxc = -1.0f + step * (float)ixv;
    const float wyc = -1.0f + step * (float)iyv;
    const float wzc = -1.0f + step * (float)izv;

    // B (4x16): VGPR0 = {K0 rows lanes 0-15 | K2 rows lanes 16-31},
    //           VGPR1 = {K1 | K3}  => lane<16 supplies (x,y), lane>=16 (z,1)
    v2f B;
    B.x = half ? wzc : wxc;
    B.y = half ? 1.0f : wyc;

    // A (16x4): lanes 0-15 give M=lane K=(0,1); lanes 16-31 give M=lane-16 K=(2,3)
    const int r  = col;
    const int k0 = half * 2;
    v2f A1, A2;
    A1.x = s1_elem(Pm, r, k0); A1.y = s1_elem(Pm, r, k0 + 1);
    A2.x = s2_elem(Pm, r, k0); A2.y = s2_elem(Pm, r, k0 + 1);

    v8f zero = {0.f, 0.f, 0.f, 0.f, 0.f, 0.f, 0.f, 0.f};
    v8f d1 = __builtin_amdgcn_wmma_f32_16x16x4_f32(false, A1, false, B,
                                                   (short)0, zero, false, false);
    v8f d2 = __builtin_amdgcn_wmma_f32_16x16x4_f32(false, A2, false, B,
                                                   (short)0, zero, false, false);

    float comp[9];
    comp[0] = d1[0]; comp[1] = d1[1]; comp[2] = d1[2];
    comp[3] = d1[3]; comp[4] = d1[4]; comp[5] = d1[5];
    comp[6] = d2[0]; comp[7] = d2[1]; comp[8] = d2[2];

    float s[NC]; float q[NC];
    #pragma unroll
    for (int c = 0; c < NC; ++c) { s[c] = 0.0f; q[c] = 0.0f; }
    float msum = 0.0f;

    // reciprocal constants for the [-1,1] normalization (replaces IEEE divides)
    const float inv_hw = 1.0f / ((float)(IMW - 1) * 0.5f);
    const float inv_hh = 1.0f / ((float)(IMH - 1) * 0.5f);

    #pragma unroll
    for (int j = 0; j < 3; ++j) {
        const float uz = comp[3 * j + 0];
        const float vz = comp[3 * j + 1];
        const float zz = comp[3 * j + 2];
        const int view = (j < 2) ? (2 * half + j) : (4 + half);

        const float zi  = __builtin_amdgcn_rcpf(zz + 1e-8f);  // v_rcp_f32, 1 ulp
        const float xy0 = uz * zi;
        const float xy1 = vz * zi;
        const float nx  = xy0 * inv_hw - 1.0f;
        const float ny  = xy1 * inv_hh - 1.0f;
        const bool  m   = (fabsf(nx) <= 1.0f) && (fabsf(ny) <= 1.0f) && (zz > 0.0f);
        if (m) {
            // align_corners=True bilinear; mask guarantees ixf in [0, W-1],
            // iyf in [0, H-1].
            const float ixf = (nx + 1.0f) * 0.5f * (float)(IMW - 1);
            const float iyf = (ny + 1.0f) * 0.5f * (float)(IMH - 1);
            const float fx0 = floorf(ixf), fy0 = floorf(iyf);
            const float wx1 = ixf - fx0,   wy1 = iyf - fy0;
            const float wx0 = 1.0f - wx1,  wy0 = 1.0f - wy1;
            int x0 = (int)fx0; x0 = x0 < 0 ? 0 : (x0 > IMW - 1 ? IMW - 1 : x0);
            int y0 = (int)fy0; y0 = y0 < 0 ? 0 : (y0 > IMH - 1 ? IMH - 1 : y0);
            const int y1 = (y0 + 1 > IMH - 1) ? IMH - 1 : y0 + 1;
            // 64-bit row-pair gather: base clamped so [xb, xb+1] is in-bounds.
            // When x0 == W-1 the reference weight wx1 is exactly 0, so folding
            // the shift into column weights (wa, wb) is exact:
            const bool hi  = (x0 > IMW - 2);
            const int  xb  = hi ? (IMW - 2) : x0;
            const float wa = hi ? 0.0f : wx0;   // weight of loaded .x
            const float wb = hi ? wx0  : wx1;   // weight of loaded .y
            const int o0 = y0 * IMW + xb;
            const int o1 = y1 * IMW + xb;
            const float* vb = feat + (size_t)view * NC * IMH * IMW;
            msum += 1.0f;
            #pragma unroll
            for (int c = 0; c < NC; ++c) {
                const float* cb = vb + (size_t)c * IMH * IMW;
                const v2fu r0 = *reinterpret_cast<const v2fu*>(cb + o0);
                const v2fu r1 = *reinterpret_cast<const v2fu*>(cb + o1);
                const float h0 = r0.x * wa + r0.y * wb;   // row y0
                const float h1 = r1.x * wa + r1.y * wb;   // row y1
                const float val = h0 * wy0 + h1 * wy1;
                s[c] += val;
                q[c] += val * val;
            }
        }
    }

    // combine lane pair (views 0,1,4 + views 2,3,5)
    const float mt    = msum + __shfl_xor(msum, 16, 32);
    const float denom = (mt <= 0.0f) ? 1e-8f : mt;
    const float rden  = __builtin_amdgcn_rcpf(denom);   // one v_rcp_f32 total

    const size_t pp = (size_t)p;
    #pragma unroll
    for (int c = 0; c < NC; ++c) {
        const float st   = s[c] + __shfl_xor(s[c], 16, 32);
        const float qt   = q[c] + __shfl_xor(q[c], 16, 32);
        const float mean = st * rden;
        const float var  = qt * rden - mean * mean;
        if (half == 0) out[(size_t)c * NP + pp] = mean;           // mean channels
        else           out[(size_t)(NC + c) * NP + pp] = var;     // var channels
    }
    if (half == 1)
        out[(size_t)(2 * NC) * NP + pp] = (mt > 1.0f) ? 1.0f : 0.0f;  // mask vol
}

// ---------------------------------------------------------------------------
extern "C" void kernel_launch(void* const* d_in, const int* in_sizes, int n_in,
                              void* d_out, int out_size, void* d_ws, size_t ws_size,
                              hipStream_t stream) {
    const float* feat  = (const float*)d_in[0];   // (6,16,480,640) fp32
    const float* intrs = (const float*)d_in[1];   // (6,3,4) fp32
    const float* c2ws  = (const float*)d_in[2];   // (6,4,4) fp32
    float* out = (float*)d_out;                   // 33 * 96^3 fp32
    float* Pm  = (float*)d_ws;                    // 72 floats scratch

    proj_setup_kernel<<<1, 32, 0, stream>>>(intrs, c2ws, Pm);

    // 16 voxels per wave, 8 waves per block -> 128 voxels/block; 96^3/128 = 6912
    volume_kernel<<<NP / 128, 256, 0, stream>>>(feat, Pm, out);
}